// HeterogeneousGAT_39805756900142
// MI455X (gfx1250) — compile-verified
//
#include <hip/hip_runtime.h>
#include <hip/hip_bf16.h>
#include <math.h>

typedef __attribute__((ext_vector_type(2))) float v2f;
typedef __attribute__((ext_vector_type(8))) float v8f;
typedef __attribute__((ext_vector_type(4))) int   v4i;

#define GAS __attribute__((address_space(1)))
#define LAS __attribute__((address_space(3)))

#define BB 2
#define LL 256
#define DD 512
#define HH 8
#define DV 64
#define NEG_INF 1.0e12f   // reference uses 1e12, not IEEE inf

#if __has_builtin(__builtin_amdgcn_global_load_async_to_lds_b128) && \
    __has_builtin(__builtin_amdgcn_s_wait_asynccnt)
#define USE_ASYNC_COPY 1
#else
#define USE_ASYNC_COPY 0
#endif

// ---------------------------------------------------------------------------
// K1: fold w_rel into W_relation:  wr2[k][h] = sum_d Wrel[k, h*64+d]*wrel[h,d]
// ---------------------------------------------------------------------------
__global__ void k_fold_wrel(const float* __restrict__ Wrel,
                            const float* __restrict__ wrel,
                            float* __restrict__ wr2) {
  int t = blockIdx.x * blockDim.x + threadIdx.x;   // 4096 threads
  int k = t >> 3, h = t & 7;
  const float* wrow = Wrel + k * DD + h * DV;
  const float* vrow = wrel + h * DV;
  float s = 0.f;
#pragma unroll 8
  for (int d = 0; d < DV; ++d) s += wrow[d] * vrow[d];
  wr2[k * HH + h] = s;
}

// ---------------------------------------------------------------------------
// K2: value = inp @ W_value   (M=512, N=512, K=512) fp32 WMMA 16x16x4
// grid (M/16, N/128), block 256 = 8 waves, wave w owns n-tile w
// ---------------------------------------------------------------------------
__global__ void k_value_gemm(const float* __restrict__ A,
                             const float* __restrict__ Wv,
                             float* __restrict__ value) {
  int wave = threadIdx.x >> 5;
  int lane = threadIdx.x & 31;
  int half = lane >> 4;        // lanes 16-31 hold K+2,K+3
  int l16  = lane & 15;
  int mBase = blockIdx.x * 16;
  int nBase = blockIdx.y * 128 + wave * 16;
  const float* arow = A + (mBase + l16) * DD;
  v8f c = {};
  for (int k0 = 0; k0 < DD; k0 += 4) {
    int ka = k0 + half * 2;
    v2f a; a.x = arow[ka]; a.y = arow[ka + 1];
    v2f b; b.x = Wv[ka * DD + nBase + l16];
           b.y = Wv[(ka + 1) * DD + nBase + l16];
    c = __builtin_amdgcn_wmma_f32_16x16x4_f32(false, a, false, b,
                                              (short)0, c, false, false);
  }
#pragma unroll
  for (int r = 0; r < 8; ++r)
    value[(mBase + r + half * 8) * DD + nBase + l16] = c[r];
}

// ---------------------------------------------------------------------------
// K3: s_src[b,h,j] = <value[b,h,j,:], w_src[h,:]> ; same for s_tgt
// ---------------------------------------------------------------------------
__global__ void k_svec(const float* __restrict__ value,
                       const float* __restrict__ wsrc,
                       const float* __restrict__ wtgt,
                       float* __restrict__ s_src,
                       float* __restrict__ s_tgt) {
  int t = blockIdx.x * 256 + threadIdx.x;   // 8192 threads
  int sel = t >> 12;
  int rem = t & 4095;
  int b = rem >> 11, h = (rem >> 8) & 7, j = rem & 255;
  const float* w = (sel ? wtgt : wsrc) + h * DV;
  const float* v = value + (b * LL + j) * DD + h * DV;
  float s = 0.f;
#pragma unroll 8
  for (int d = 0; d < DV; ++d) s += v[d] * w[d];
  (sel ? s_tgt : s_src)[(b * HH + h) * LL + j] = s;
}

// ---------------------------------------------------------------------------
// K4: s_rel = relation_inp(131072 x 512) @ wr2(512 x 8, padded N=16)
// THE bandwidth kernel: streams 268 MB once.
// block = 128 thr = 4 waves, wave = one 16-row tile.
// Async path: per-wave double-buffered global->LDS staging via
// global_load_async_to_lds_b128 (512B fully-coalesced per instr), ASYNCcnt
// pipelined, conflict-free ds_load_b64 feeds WMMA fragments.
// ---------------------------------------------------------------------------
#define KC   64      // K-chunk (floats) staged per buffer
#define AROW 68      // padded LDS row stride: 272B, 16B-aligned, 4-bank stride

__global__ void k_srel(const float* __restrict__ rel,
                       const float* __restrict__ wr2,
                       float* __restrict__ s_rel) {
  __shared__ float lw[DD * HH];                 // 16 KB folded weights
  for (int i = threadIdx.x; i < DD * HH; i += 128) lw[i] = wr2[i];
  __syncthreads();

  int wave = threadIdx.x >> 5;
  int lane = threadIdx.x & 31;
  int half = lane >> 4;
  int l16  = lane & 15;
  long tile    = (long)blockIdx.x * 4 + wave;
  long rowBase = tile * 16;                     // row in flattened (b,i,j)
  int j0 = (int)(rowBase & 255);                // 16-row tile never crosses j wrap
  int i_ = (int)((rowBase >> 8) & 255);
  int b  = (int)(rowBase >> 16);
  int  nsel   = l16 & 7;
  bool nvalid = l16 < 8;
  v8f c = {};

#if USE_ASYNC_COPY
  __shared__ __align__(16) float atile[2][4][16 * AROW];   // ~34 KB
  const float* gbase = rel + rowBase * DD;
  float* lb0 = &atile[0][wave][0];
  float* lb1 = &atile[1][wave][0];

  // one chunk = 16 rows x 64 floats; 8 async b128 instrs, 512B coalesced each
  auto issue_chunk = [&](int kc, float* lb) {
#pragma unroll
    for (int it = 0; it < 8; ++it) {
      int row = it * 2 + half;                  // lanes 0-15: row it*2, 16-31: it*2+1
      const float* g = gbase + row * DD + kc + l16 * 4;
      float* l = lb + row * AROW + l16 * 4;
      __builtin_amdgcn_global_load_async_to_lds_b128(
          (GAS v4i*)g, (LAS v4i*)l, 0, 0);
    }
  };

  issue_chunk(0, lb0);
  for (int ch = 0; ch < DD / KC; ++ch) {        // 8 chunks
    float* lb = (ch & 1) ? lb1 : lb0;
    if (ch < DD / KC - 1) {
      issue_chunk((ch + 1) * KC, (ch & 1) ? lb0 : lb1);
      __builtin_amdgcn_s_wait_asynccnt(8);      // in-order: chunk ch complete
    } else {
      __builtin_amdgcn_s_wait_asynccnt(0);
    }
    int kcBase = ch * KC;
#pragma unroll 4
    for (int kk = 0; kk < KC; kk += 4) {
      int ka = kk + half * 2;
      v2f a = *(const v2f*)(lb + l16 * AROW + ka);   // ds_load_b64, all 64 banks
      int kg = kcBase + ka;
      float b0 = lw[kg * HH + nsel];
      float b1 = lw[(kg + 1) * HH + nsel];
      v2f bb; bb.x = nvalid ? b0 : 0.f; bb.y = nvalid ? b1 : 0.f;
      c = __builtin_amdgcn_wmma_f32_16x16x4_f32(false, a, false, bb,
                                                (short)0, c, false, false);
    }
  }
#else
  const float* arow = rel + (rowBase + l16) * DD;
  for (int k0 = 0; k0 < DD; k0 += 4) {
    int ka = k0 + half * 2;
    v2f a = *(const v2f*)(arow + ka);           // 8B aligned
    float b0 = lw[ka * HH + nsel];
    float b1 = lw[(ka + 1) * HH + nsel];
    v2f bb; bb.x = nvalid ? b0 : 0.f; bb.y = nvalid ? b1 : 0.f;
    c = __builtin_amdgcn_wmma_f32_16x16x4_f32(false, a, false, bb,
                                              (short)0, c, false, false);
  }
#endif

  if (nvalid) {
    int h = l16;
    float* outp = s_rel + ((long)(b * HH + h) * LL + i_) * LL + j0;
#pragma unroll
    for (int r = 0; r < 8; ++r) outp[r + half * 8] = c[r];
  }
}

// ---------------------------------------------------------------------------
// K5: scores -> leaky -> masks -> softmax -> renorm -> out = attn@value + inp
// block per (b,h,iTile of 16 rows); 256 threads = 16 rows x 16 cols
// ---------------------------------------------------------------------------
__global__ void k_attn_out(const float* __restrict__ s_src,
                           const float* __restrict__ s_tgt,
                           const float* __restrict__ s_rel,
                           const unsigned char* __restrict__ mask,
                           const float* __restrict__ adj,
                           const float* __restrict__ value,
                           const float* __restrict__ inp,
                           float* __restrict__ out) {
  const int AS = 258;                           // padded stride: bank-safe, 8B aligned
  __shared__ float att[16 * AS];                // attn rows (16 x 256)
  __shared__ float red[16][17];
  __shared__ float rowval[16];

  int blk   = blockIdx.x;
  int iTile = blk & 15;
  int h     = (blk >> 4) & 7;
  int b     = blk >> 7;
  int iBase = iTile * 16;
  int r   = threadIdx.x >> 4;
  int c16 = threadIdx.x & 15;
  int i   = iBase + r;
  int bh  = b * HH + h;

  const float* srel_row = s_rel + ((long)bh * LL + i) * LL;
  const float* ssrc = s_src + bh * LL;
  float stg = s_tgt[bh * LL + i];
  const unsigned char* mrow = mask + ((long)b * LL + i) * LL;
  const float* arow = adj + ((long)bh * LL + i) * LL;

  float sco[16], adjv[16];
  unsigned int mbits = 0;
  float lmax = -3.4e38f;
#pragma unroll
  for (int q = 0; q < 16; ++q) {
    int j = c16 * 16 + q;
    float s = ssrc[j] + stg + srel_row[j];
    s = (s >= 0.f) ? s : 0.2f * s;              // leaky relu before masking
    float av = arow[j];
    adjv[q] = av;
    bool mm = mrow[j] != 0;
    if (mm) mbits |= (1u << q);
    if (mm || av == 0.f) s = -NEG_INF;
    sco[q] = s;
    lmax = fmaxf(lmax, s);
  }
  red[r][c16] = lmax;
  __syncthreads();
  if (c16 == 0) {
    float m = red[r][0];
    for (int t = 1; t < 16; ++t) m = fmaxf(m, red[r][t]);
    rowval[r] = m;
  }
  __syncthreads();
  float rmax = rowval[r];
  float lsum = 0.f;
#pragma unroll
  for (int q = 0; q < 16; ++q) {
    float e = __expf(sco[q] - rmax);
    sco[q] = e; lsum += e;
  }
  __syncthreads();
  red[r][c16] = lsum;
  __syncthreads();
  if (c16 == 0) {
    float s = 0.f;
    for (int t = 0; t < 16; ++t) s += red[r][t];
    rowval[r] = s;
  }
  __syncthreads();
  float rsum = rowval[r];
  float labs = 0.f;
#pragma unroll
  for (int q = 0; q < 16; ++q) {
    float a = sco[q] / rsum;
    float inv = (adjv[q] == 0.f) ? 1e-12f : 1.f / adjv[q];
    a *= inv;
    sco[q] = a; labs += fabsf(a);
  }
  __syncthreads();
  red[r][c16] = labs;
  __syncthreads();
  if (c16 == 0) {
    float s = 0.f;
    for (int t = 0; t < 16; ++t) s += red[r][t];
    rowval[r] = fmaxf(s, 1e-12f);
  }
  __syncthreads();
  float rabs = rowval[r];
#pragma unroll
  for (int q = 0; q < 16; ++q) {
    float a = sco[q] / rabs;
    if ((mbits >> q) & 1) a = 0.f;
    if (adjv[q] == 0.f) a = 0.f;
    att[r * AS + c16 * 16 + q] = a;
  }
  __syncthreads();

  // phase 2: out tile (16 x 64) = att(16 x 256) @ value(256 x 64), 4 WMMA waves
  int wave = threadIdx.x >> 5;
  if (wave < 4) {
    int lane = threadIdx.x & 31;
    int half = lane >> 4;
    int l16  = lane & 15;
    int nBase = wave * 16;
    const float* vcol = value + ((long)b * LL) * DD + h * DV + nBase + l16;
    v8f c = {};
    for (int k0 = 0; k0 < LL; k0 += 4) {
      int ka = k0 + half * 2;
      v2f a; a.x = att[l16 * AS + ka]; a.y = att[l16 * AS + ka + 1];
      v2f bb; bb.x = vcol[(long)ka * DD]; bb.y = vcol[(long)(ka + 1) * DD];
      c = __builtin_amdgcn_wmma_f32_16x16x4_f32(false, a, false, bb,
                                                (short)0, c, false, false);
    }
#pragma unroll
    for (int rr = 0; rr < 8; ++rr) {
      long oidx = ((long)b * LL + iBase + rr + half * 8) * DD + h * DV + nBase + l16;
      out[oidx] = c[rr] + inp[oidx];            // fused residual add
    }
  }
}

// ---------------------------------------------------------------------------
extern "C" void kernel_launch(void* const* d_in, const int* in_sizes, int n_in,
                              void* d_out, int out_size, void* d_ws, size_t ws_size,
                              hipStream_t stream) {
  const float* inp  = (const float*)d_in[0];
  const float* rel  = (const float*)d_in[1];
  const unsigned char* mask = (const unsigned char*)d_in[2];
  const float* adj  = (const float*)d_in[3];
  const float* Wv   = (const float*)d_in[4];
  const float* Wr   = (const float*)d_in[5];
  const float* wsrc = (const float*)d_in[6];
  const float* wtgt = (const float*)d_in[7];
  const float* wrel = (const float*)d_in[8];
  float* out = (float*)d_out;

  float* ws    = (float*)d_ws;
  float* wr2   = ws;                  // 4096
  float* value = wr2 + 4096;          // 262144
  float* s_src = value + 262144;      // 4096
  float* s_tgt = s_src + 4096;        // 4096
  float* s_rel = s_tgt + 4096;        // 1048576  (~5.3 MB total)

  k_fold_wrel<<<16, 256, 0, stream>>>(Wr, wrel, wr2);
  k_value_gemm<<<dim3(32, 4), 256, 0, stream>>>(inp, Wv, value);
  k_svec<<<32, 256, 0, stream>>>(value, wsrc, wtgt, s_src, s_tgt);
  k_srel<<<2048, 128, 0, stream>>>(rel, wr2, s_rel);
  k_attn_out<<<256, 256, 0, stream>>>(s_src, s_tgt, s_rel, mask, adj,
                                      value, inp, out);
}